// LecunLCN_89850715832862
// MI455X (gfx1250) — compile-verified
//
#include <hip/hip_runtime.h>
#include <math.h>
#include <stdint.h>

// ---------------------------------------------------------------------------
// LeCun LCN, fused single kernel for MI455X (gfx1250, wave32).
// Both separable 19-tap Gaussian blurs are executed as banded-Toeplitz GEMMs
// on V_WMMA_F32_16X16X4_F32 (exact fp32 math, 2048 flop/instr).
// X tile is filled with GLOBAL_LOAD_ASYNC_TO_LDS_B32 (ASYNCcnt path).
// ---------------------------------------------------------------------------

typedef __attribute__((ext_vector_type(2))) float v2f;
typedef __attribute__((ext_vector_type(8))) float v8f;

#define IMG_H 512
#define IMG_W 512
#define NCH   3
#define TILE  64

// LDS plane geometry (one channel per workgroup). Logical origins:
//  X    rows/cols start at -18  (input + halo + WMMA K-slack)
//  H1   rows start -18, cols start -9   (stage-1 horizontal blur)
//  CE   rows/cols start -9              (centered = x - mean)
//  H2   rows start -9, cols start 0     (stage-2 horizontal blur of cent^2)
#define XP  117
#define XR  112
#define XC  116
#define H1P 97
#define H1R 116
#define CP  85
#define CR  96
#define H2P 65
#define H2R 84

#define X_OFF  0
#define H1_OFF (XR * XP)                 // 13104
#define C_OFF  (H1_OFF + H1R * H1P)      // +11252
#define H2_OFF (C_OFF + CR * CP)         // +8160
#define LDS_TOT (H2_OFF + H2R * H2P)     // 37976 floats = 151,904 B (<320KB/WGP)

struct Taps { float g[19]; };

static __device__ inline v8f wmma4(v2f a, v2f b, v8f c) {
  // D(16x16,f32) = A(16x4,f32) * B(4x16,f32) + C ; exact fp32 path
  return __builtin_amdgcn_wmma_f32_16x16x4_f32(
      /*neg_a=*/false, a, /*neg_b=*/false, b,
      /*c_mod=*/(short)0, c, /*reuse_a=*/false, /*reuse_b=*/false);
}

__global__ __launch_bounds__(512) void lcn_kernel(const float* __restrict__ x0,
                                                  const float* __restrict__ mask,
                                                  float* __restrict__ out,
                                                  Taps T) {
  __shared__ float lds[LDS_TOT];
  float* X  = lds + X_OFF;
  float* H1 = lds + H1_OFF;
  float* CE = lds + C_OFF;
  float* H2 = lds + H2_OFF;

  const int tid  = threadIdx.x;
  const int lane = tid & 31;
  const int wave = tid >> 5;
  const int q    = lane & 15;    // M (A) or N (B) index of this lane
  const int half = lane >> 4;    // K sub-pair selector

  const int b    = blockIdx.z / NCH;
  const int ch   = blockIdx.z % NCH;
  const int row0 = blockIdx.y * TILE;
  const int col0 = blockIdx.x * TILE;

  // Banded-Toeplitz operand, resident in 18 VGPRs for the whole kernel.
  // f32 A(16x4) and B(4x16) lane layouts are transpose-symmetric, so the SAME
  // registers act as A in vertical passes and B in horizontal passes:
  //   value = g[r - q],  r = kb*4 + 2*half + i  (zero outside the 19-tap band)
  v2f toe[9];
#pragma unroll
  for (int kb = 0; kb < 9; ++kb) {
#pragma unroll
    for (int i = 0; i < 2; ++i) {
      int r = kb * 4 + 2 * half + i;
      int d = r - q;
      toe[kb][i] = (d >= 0 && d <= 18) ? T.g[d] : 0.0f;
    }
  }

  // Zero-init H1 / CE / H2 (edge-tile WMMA operand reads hit zeroed slack).
  for (int idx = tid; idx < LDS_TOT - H1_OFF; idx += 512) lds[H1_OFF + idx] = 0.0f;

  // Load X tile with halo via async global->LDS DMA; SAME conv semantics ->
  // zero outside the image (divergent else-path writes zeros through DS).
  for (int idx = tid; idx < XR * XC; idx += 512) {
    int r = idx / XC, c = idx % XC;
    int gr = row0 + r - 18, gc = col0 + c - 18;
    float* lp = X + r * XP + c;
    if (gr >= 0 && gr < IMG_H && gc >= 0 && gc < IMG_W) {
      const float* gp = x0 + ((((size_t)b * IMG_H + gr) * IMG_W + gc) * NCH + ch);
      unsigned lds_addr = (unsigned)(uintptr_t)lp;           // low 32b = LDS offset
      unsigned long long ga = (unsigned long long)(uintptr_t)gp;
      asm volatile("global_load_async_to_lds_b32 %0, %1, off"
                   :: "v"(lds_addr), "v"(ga) : "memory");
    } else {
      *lp = 0.0f;
    }
  }
  asm volatile("s_wait_asynccnt 0x0" ::: "memory");
  __syncthreads();

  // ---- Stage 1 H: H1[r][c] = sum_j g[j] X[r][c+j-9] ; 7x6 tiles -----------
  for (int t = wave; t < 42; t += 16) {
    int r0 = -18 + (t / 6) * 16;
    int c0 = -9  + (t % 6) * 16;
    const float* ap = X + (r0 + 18 + q) * XP + (c0 + 9) + 2 * half;  // c0-9+18
    v8f acc = {};
#pragma unroll
    for (int kb = 0; kb < 9; ++kb) {
      v2f a; a[0] = ap[0]; a[1] = ap[1]; ap += 4;
      acc = wmma4(a, toe[kb], acc);            // data x Toeplitz
    }
    int cl = c0 + q;
#pragma unroll
    for (int i = 0; i < 8; ++i) {
      int rl = r0 + 8 * half + i;
      if (rl < 82 && cl < 73) H1[(rl + 18) * H1P + (cl + 9)] = acc[i];
    }
  }
  __syncthreads();

  // ---- Stage 1 V: mean = sum_j g[j] H1[r+j-9][c] ; centered = x - mean ----
  for (int t = wave; t < 36; t += 16) {
    int r0 = -9 + (t / 6) * 16;
    int c0 = -9 + (t % 6) * 16;
    const float* bp = H1 + (r0 + 9 + 2 * half) * H1P + (c0 + 9 + q); // r0-9+18
    v8f acc = {};
#pragma unroll
    for (int kb = 0; kb < 9; ++kb) {
      v2f bv; bv[0] = bp[0]; bv[1] = bp[H1P]; bp += 4 * H1P;
      acc = wmma4(toe[kb], bv, acc);           // Toeplitz x data
    }
    int cl = c0 + q;
    int gc = col0 + cl;
#pragma unroll
    for (int i = 0; i < 8; ++i) {
      int rl = r0 + 8 * half + i;
      int gr = row0 + rl;
      // centered is zero outside the image (SAME zero-padding for blur #2)
      if (rl < 73 && cl < 73 && gr >= 0 && gr < IMG_H && gc >= 0 && gc < IMG_W) {
        float xv = X[(rl + 18) * XP + (cl + 18)];
        CE[(rl + 9) * CP + (cl + 9)] = xv - acc[i];
      }
    }
  }
  __syncthreads();

  // ---- Stage 2 H: H2[r][c] = sum_j g[j] CE[r][c+j-9]^2 ; 6x4 tiles --------
  for (int t = wave; t < 24; t += 16) {
    int r0 = -9 + (t / 4) * 16;
    int c0 = (t % 4) * 16;
    const float* ap = CE + (r0 + 9 + q) * CP + c0 + 2 * half;        // c0-9+9
    v8f acc = {};
#pragma unroll
    for (int kb = 0; kb < 9; ++kb) {
      float u = ap[0], v = ap[1]; ap += 4;
      v2f a; a[0] = u * u; a[1] = v * v;       // square on A-load (no SQ buf)
      acc = wmma4(a, toe[kb], acc);
    }
    int cl = c0 + q;
#pragma unroll
    for (int i = 0; i < 8; ++i) {
      int rl = r0 + 8 * half + i;
      if (rl < 73) H2[(rl + 9) * H2P + cl] = acc[i];
    }
  }
  __syncthreads();

  // ---- Stage 2 V + finalize: var -> out = cent/(sqrt(var)+eps)*mask ------
  for (int t = wave; t < 16; t += 16) {
    int r0 = (t / 4) * 16;
    int c0 = (t % 4) * 16;
    const float* bp = H2 + (r0 + 2 * half) * H2P + (c0 + q);         // r0-9+9
    v8f acc = {};
#pragma unroll
    for (int kb = 0; kb < 9; ++kb) {
      v2f bv; bv[0] = bp[0]; bv[1] = bp[H2P]; bp += 4 * H2P;
      acc = wmma4(toe[kb], bv, acc);
    }
    int cl = c0 + q;
    int gc = col0 + cl;
#pragma unroll
    for (int i = 0; i < 8; ++i) {
      int rl = r0 + 8 * half + i;
      int gr = row0 + rl;
      float ce = CE[(rl + 9) * CP + (cl + 9)];
      size_t pix = ((size_t)b * IMG_H + gr) * IMG_W + gc;
      float m = mask[pix];
      out[pix * NCH + ch] = ce / (sqrtf(acc[i]) + 1.0e-4f) * m;
    }
  }
}

extern "C" void kernel_launch(void* const* d_in, const int* in_sizes, int n_in,
                              void* d_out, int out_size, void* d_ws, size_t ws_size,
                              hipStream_t stream) {
  const float* x0   = (const float*)d_in[0];
  const float* mask = (const float*)d_in[1];
  float* out = (float*)d_out;

  // Normalized 19-tap 1-D Gaussian (outer-product kernel is separable; the
  // 2-D normalization factors into v/sum(v) per axis). Computed on host.
  Taps T;
  double v[19], s = 0.0;
  const double inv_sqrt2pi = 0.3989422804014327;  // 1/sqrt(2*pi)
  for (int t = 0; t < 19; ++t) {
    double d = (double)(t - 9);
    v[t] = exp(-0.5 * d * d) * inv_sqrt2pi;
    s += v[t];
  }
  for (int t = 0; t < 19; ++t) T.g[t] = (float)(v[t] / s);

  const int B = in_sizes[0] / (IMG_H * IMG_W * NCH);   // 32
  dim3 grid(IMG_W / TILE, IMG_H / TILE, B * NCH);
  lcn_kernel<<<grid, 512, 0, stream>>>(x0, mask, out, T);
}